// VAMPBlock_14551349199047
// MI455X (gfx1250) — compile-verified
//
#include <hip/hip_runtime.h>

// ---------------------------------------------------------------------------
// GCN layer + MLP for MI455X (gfx1250, wave32).
// Precision: fp32 end-to-end -> native V_WMMA_F32_16X16X4_F32 matrix ops.
// Problem is HBM/scatter bound (~0.6GB random edge traffic @ 23.3 TB/s),
// so fp32 WMMA is free; weights are LDS-resident and reused per block.
// ---------------------------------------------------------------------------

#define FDIM 128   // DIN == DOUT == 128
#define HDIM 256   // DH

typedef __attribute__((ext_vector_type(2))) float v2f;
typedef __attribute__((ext_vector_type(8))) float v8f;

// ---------------------------------------------------------------------------
// C[M,NC] = act_out( act_in(A)[M,K] @ W[K,NC] + bias )
// Block: 256 threads = 8 wave32; wave w owns rows [blk*128 + 16w, +16).
// W streamed through LDS in 64-row chunks (<= 64KB static LDS).
// WMMA f32 16x16x4 fragment layouts (wave32):
//   A (16x4): lane L -> row L%16 ; VGPR v -> K = v + 2*(L/16)
//   B (4x16): lane L -> col L%16 ; VGPR v -> K = v + 2*(L/16)
//   C (16x16): lane L -> col L%16 ; VGPR v -> row = v + 8*(L/16)
// ---------------------------------------------------------------------------
template <int K, int NC, bool RELU_IN, bool RELU_OUT>
__global__ __launch_bounds__(256) void gemm_wmma_f32(
    const float* __restrict__ A, const float* __restrict__ W,
    const float* __restrict__ bias, float* __restrict__ C, int M) {
  constexpr int KC = 64;           // K-chunk rows of W staged in LDS
  constexpr int NTILES = NC / 16;  // 16x16 output tiles per wave
  __shared__ float sW[KC * NC];

  const int tid = threadIdx.x;
  const int wave = tid >> 5;
  const int lane = tid & 31;
  const int lhalf = lane >> 4;  // 0 | 1
  const int lmod = lane & 15;

  const int m0 = blockIdx.x * 128 + wave * 16;
  const bool active = (m0 < M);  // uniform per wave; M % 16 == 0

  // Seed accumulators with bias (bias depends only on column = lane%16).
  v8f acc[NTILES];
#pragma unroll
  for (int j = 0; j < NTILES; ++j) {
    const float bv = bias[j * 16 + lmod];
#pragma unroll
    for (int v = 0; v < 8; ++v) acc[j][v] = bv;
  }

  const float* Arow = A + (long)(m0 + lmod) * K;  // this lane's A row

  for (int chunk = 0; chunk < K / KC; ++chunk) {
    // Cooperative 128-bit loads of the KC x NC weight chunk into LDS.
    const float4* Wg = (const float4*)(W + (long)chunk * KC * NC);
    float4* Ws = (float4*)sW;
    for (int i = tid; i < (KC * NC) / 4; i += 256) Ws[i] = Wg[i];
    __syncthreads();

    if (active) {
      for (int k = 0; k < KC; k += 4) {
        // A fragment: two consecutive K values -> one 8-byte load.
        const int kk = chunk * KC + k + 2 * lhalf;
        const float2 av = *(const float2*)(Arow + kk);
        v2f a;
        a[0] = RELU_IN ? fmaxf(av.x, 0.0f) : av.x;
        a[1] = RELU_IN ? fmaxf(av.y, 0.0f) : av.y;
#pragma unroll
        for (int j = 0; j < NTILES; ++j) {
          v2f b;
          b[0] = sW[(k + 2 * lhalf + 0) * NC + j * 16 + lmod];
          b[1] = sW[(k + 2 * lhalf + 1) * NC + j * 16 + lmod];
          acc[j] = __builtin_amdgcn_wmma_f32_16x16x4_f32(
              false, a, false, b, (short)0, acc[j], false, false);
        }
      }
    }
    __syncthreads();
  }

  if (active) {
#pragma unroll
    for (int j = 0; j < NTILES; ++j) {
#pragma unroll
      for (int v = 0; v < 8; ++v) {
        float val = acc[j][v];
        if (RELU_OUT) val = fmaxf(val, 0.0f);
        C[(long)(m0 + v + 8 * lhalf) * NC + j * 16 + lmod] = val;
      }
    }
  }
}

// --------------------------- graph aggregation -----------------------------

__global__ void deg_init_kernel(float* deg, int n) {
  int i = blockIdx.x * blockDim.x + threadIdx.x;
  if (i < n) deg[i] = 1.0f;  // self loop contributes 1
}

__global__ void deg_accum_kernel(const long long* __restrict__ rows,
                                 float* deg, int e) {
  int i = blockIdx.x * blockDim.x + threadIdx.x;
  if (i < e) atomicAdd(&deg[(int)rows[i]], 1.0f);
}

__global__ void deg_rsqrt_kernel(float* deg, int n) {
  int i = blockIdx.x * blockDim.x + threadIdx.x;
  if (i < n) deg[i] = rsqrtf(deg[i]);  // deg buffer now holds dis = deg^-1/2
}

// Self-loop term: m[i,:] = xlin[i,:] * dis[i]^2
__global__ void self_msg_kernel(const float* __restrict__ xlin,
                                const float* __restrict__ dis,
                                float* __restrict__ m, long total) {
  long idx = (long)blockIdx.x * blockDim.x + threadIdx.x;
  if (idx < total) {
    int node = (int)(idx >> 7);  // /128
    float d = dis[node];
    m[idx] = xlin[idx] * d * d;
  }
}

// Edge scatter: 128 threads per edge (feature-parallel, coalesced rows).
__global__ void edge_scatter_kernel(const long long* __restrict__ rows,
                                    const long long* __restrict__ cols,
                                    const float* __restrict__ dis,
                                    const float* __restrict__ xlin,
                                    float* __restrict__ m, int e) {
  int eidx = blockIdx.x * 2 + (threadIdx.x >> 7);
  int f = threadIdx.x & 127;
  if (eidx < e) {
    int r = (int)rows[eidx];
    int c = (int)cols[eidx];
    float w = dis[r] * dis[c];
    atomicAdd(&m[(long)r * FDIM + f], xlin[(long)c * FDIM + f] * w);
  }
}

// --------------------------------- driver ----------------------------------

extern "C" void kernel_launch(void* const* d_in, const int* in_sizes, int n_in,
                              void* d_out, int out_size, void* d_ws,
                              size_t ws_size, hipStream_t stream) {
  const float* x = (const float*)d_in[0];
  const long long* eidx = (const long long*)d_in[1];  // int64 [2, E]
  const float* lin_w = (const float*)d_in[2];
  const float* lin_b = (const float*)d_in[3];
  const float* w1 = (const float*)d_in[4];
  const float* b1 = (const float*)d_in[5];
  const float* w2 = (const float*)d_in[6];
  const float* b2 = (const float*)d_in[7];

  const int n = in_sizes[0] / FDIM;  // 50000
  const int e = in_sizes[1] / 2;     // 600000
  const long long* rows = eidx;
  const long long* cols = eidx + e;

  // Workspace layout (fp32): deg/dis | xlin | m | h1  (~103 MB total)
  float* deg = (float*)d_ws;
  float* xlin = deg + n;
  float* m = xlin + (size_t)n * FDIM;
  float* h1 = m + (size_t)n * FDIM;
  float* out = (float*)d_out;

  const int gemm_blocks = (n + 127) / 128;

  // 1) degrees -> dis = rsqrt(deg)
  deg_init_kernel<<<(n + 255) / 256, 256, 0, stream>>>(deg, n);
  deg_accum_kernel<<<(e + 255) / 256, 256, 0, stream>>>(rows, deg, e);
  deg_rsqrt_kernel<<<(n + 255) / 256, 256, 0, stream>>>(deg, n);

  // 2) x_lin = x @ lin_w + lin_b
  gemm_wmma_f32<FDIM, FDIM, false, false>
      <<<gemm_blocks, 256, 0, stream>>>(x, lin_w, lin_b, xlin, n);

  // 3) normalized aggregation: self loops then edge scatter-add
  {
    long total = (long)n * FDIM;
    self_msg_kernel<<<(int)((total + 255) / 256), 256, 0, stream>>>(xlin, deg,
                                                                    m, total);
    edge_scatter_kernel<<<(e + 1) / 2, 256, 0, stream>>>(rows, cols, deg, xlin,
                                                         m, e);
  }

  // 4) h1 = relu(relu(m) @ w1 + b1)   (input ReLU fuses z = relu(m))
  gemm_wmma_f32<FDIM, HDIM, true, true>
      <<<gemm_blocks, 256, 0, stream>>>(m, w1, b1, h1, n);

  // 5) out = h1 @ w2 + b2
  gemm_wmma_f32<HDIM, FDIM, false, false>
      <<<gemm_blocks, 256, 0, stream>>>(h1, w2, b2, out, n);
}